// NodeModelIn_32796370272848
// MI455X (gfx1250) — compile-verified
//
#include <hip/hip_runtime.h>

#define N_NODES 100000
#define N_EDGES 1600000
#define D_EDGE  32
#define OUT_D   96   // [mean(32) | max(32) | sum(32)]

// ---------------------------------------------------------------------------
// Kernel 1: initialize accumulators.
//   out[row, 32..63] = -inf   (max accumulator, valid init for maximumNumber)
//   out[row, 64..95] = 0      (sum accumulator)
//   cnt[row]         = 0
// ---------------------------------------------------------------------------
__global__ void nm_init_kernel(float* __restrict__ out,
                               unsigned int* __restrict__ cnt) {
    int tid = blockIdx.x * blockDim.x + threadIdx.x;
    if (tid < N_NODES * D_EDGE) {
        int row  = tid >> 5;
        int lane = tid & 31;
        out[row * OUT_D + 32 + lane] = __int_as_float(0xFF800000); // -inf
        out[row * OUT_D + 64 + lane] = 0.0f;
        if (lane == 0) cnt[row] = 0u;
    }
}

// ---------------------------------------------------------------------------
// Kernel 2: scatter. One wave32 per edge per iteration; lane == feature dim.
//   - attr row load: one coalesced 128B transaction per wave
//   - col index is wave-uniform (readfirstlane) -> scalar load path
//   - sum: native global_atomic_add_f32 (non-returning, L2-resident buffer)
//   - max: native global_atomic_max_num_f32 (IEEE maximumNumber) — single
//     branch-free instruction per edge, no CAS loop, no sign-split divergence
//   - count: one lane per edge bumps cnt[col]
// ---------------------------------------------------------------------------
__global__ void nm_scatter_kernel(const float* __restrict__ attr,
                                  const int*   __restrict__ col,   // edge_index row 1
                                  float*       __restrict__ out,
                                  unsigned int* __restrict__ cnt,
                                  int num_waves) {
    const int lane = threadIdx.x & 31;
    const int wave = __builtin_amdgcn_readfirstlane(
        (int)((blockIdx.x * blockDim.x + threadIdx.x) >> 5));

    for (int e = wave; e < N_EDGES; e += num_waves) {
        // speculative prefetch of the next edge row this wave will touch
        __builtin_prefetch(attr + (size_t)(e + num_waves) * D_EDGE + lane, 0, 3);

        const int   c = col[e];                          // wave-uniform
        const float v = attr[(size_t)e * D_EDGE + lane]; // coalesced 128B/wave

        float* base = out + (size_t)c * OUT_D;

        unsafeAtomicAdd(base + 64 + lane, v);   // sum: global_atomic_add_f32
        unsafeAtomicMax(base + 32 + lane, v);   // max: global_atomic_max_num_f32

        if (lane == 0) atomicAdd(cnt + c, 1u);
    }
}

// ---------------------------------------------------------------------------
// Kernel 3: finalize.
//   mean = sum / max(count,1) -> cols 0..31
//   max  = count>0 ? max : 0  -> cols 32..63 (in place, clears -inf)
//   sum stays in cols 64..95
// ---------------------------------------------------------------------------
__global__ void nm_finalize_kernel(float* __restrict__ out,
                                   const unsigned int* __restrict__ cnt) {
    int tid = blockIdx.x * blockDim.x + threadIdx.x;
    if (tid < N_NODES * D_EDGE) {
        int row  = tid >> 5;
        int lane = tid & 31;
        unsigned int c = cnt[row];
        float s = out[row * OUT_D + 64 + lane];
        float denom = (c > 0u) ? (float)c : 1.0f;
        out[row * OUT_D + lane] = s / denom;
        float m = out[row * OUT_D + 32 + lane];
        out[row * OUT_D + 32 + lane] = (c > 0u) ? m : 0.0f;
    }
}

// ---------------------------------------------------------------------------
// Launcher. Inputs: d_in[0]=x (unused), d_in[1]=edge_index (2 x N_EDGES int),
// d_in[2]=edge_attr (N_EDGES x 32 f32). Output: N_NODES x 96 f32.
// Workspace: N_NODES u32 counters (400 KB).
// ---------------------------------------------------------------------------
extern "C" void kernel_launch(void* const* d_in, const int* in_sizes, int n_in,
                              void* d_out, int out_size, void* d_ws, size_t ws_size,
                              hipStream_t stream) {
    const float* attr      = (const float*)d_in[2];
    const int*   edge_idx  = (const int*)d_in[1];
    const int*   col       = edge_idx + N_EDGES;   // row 1 of edge_index
    float*       out       = (float*)d_out;
    unsigned int* cnt      = (unsigned int*)d_ws;

    (void)in_sizes; (void)n_in; (void)out_size; (void)ws_size;

    const int threads = 256;

    // init: one thread per (node, feature)
    int init_blocks = (N_NODES * D_EDGE + threads - 1) / threads;
    nm_init_kernel<<<init_blocks, threads, 0, stream>>>(out, cnt);

    // scatter: 4096 blocks x 8 wave32 = 32768 waves, grid-stride over edges
    const int scatter_blocks = 4096;
    const int num_waves = scatter_blocks * (threads / 32);
    nm_scatter_kernel<<<scatter_blocks, threads, 0, stream>>>(attr, col, out, cnt,
                                                              num_waves);

    // finalize
    nm_finalize_kernel<<<init_blocks, threads, 0, stream>>>(out, cnt);
}